// LossFunction_43267500540639
// MI455X (gfx1250) — compile-verified
//
#include <hip/hip_runtime.h>

// Ext-vector types (WMMA operand shapes per CDNA5 ISA: A=16x4 f32 -> 2 VGPRs,
// B=4x16 f32 -> 2 VGPRs, C/D=16x16 f32 -> 8 VGPRs, wave32).
typedef float v2f   __attribute__((ext_vector_type(2)));
typedef float v8f   __attribute__((ext_vector_type(8)));
typedef float f32x4 __attribute__((ext_vector_type(4)));

// Full-wave (32-lane) sum via one V_WMMA_F32_16X16X4_F32.
// Each lane contributes s0 (A vgpr0) and s1 (A vgpr1); with B = ones(4x16),
// D[m][n] = sum_k A[m][k], so every column of D totals to the grand sum of A.
// Lane L (0..15) holds D[0..7][L] in d0..d7; lane L+16 holds D[8..15][L].
// Summing a lane's 8 regs + shfl_xor(16) yields the full 64-value sum in all lanes.
// EXEC must be all-ones at the call site (no divergence).
__device__ __forceinline__ float wave_sum_wmma(float s0, float s1) {
  v2f a; a.x = s0;   a.y = s1;
  v2f b; b.x = 1.0f; b.y = 1.0f;
  v8f c = {};
  v8f d = __builtin_amdgcn_wmma_f32_16x16x4_f32(
      /*neg_a=*/false, a, /*neg_b=*/false, b,
      /*c_mod=*/(short)0, c, /*reuse_a=*/false, /*reuse_b=*/false);
  float half_col = ((d[0] + d[1]) + (d[2] + d[3])) + ((d[4] + d[5]) + (d[6] + d[7]));
  half_col += __shfl_xor(half_col, 16, 32);  // pair lanes L and L+16 -> full sum
  return half_col;
}

__device__ __forceinline__ float masked_sq(float t, float p) {
  float d = t - p;
  return (t != -1.0f) ? d * d : 0.0f;
}

// Kernel 1: grid-stride masked-SSE over float4 lanes, one partial per block.
__global__ void __launch_bounds__(256)
masked_sse_partial(const float* __restrict__ pred,
                   const float* __restrict__ tgt,
                   float* __restrict__ partials,
                   long long n) {
  const long long nvec   = n >> 2;                                   // float4 count
  const long long tid    = (long long)blockIdx.x * blockDim.x + threadIdx.x;
  const long long stride = (long long)gridDim.x * blockDim.x;

  const f32x4* __restrict__ p4 = (const f32x4*)pred;
  const f32x4* __restrict__ t4 = (const f32x4*)tgt;

  // 4 independent accumulator chains (one per vec4 component).
  float a0 = 0.f, a1 = 0.f, a2 = 0.f, a3 = 0.f;
  for (long long i = tid; i < nvec; i += stride) {
    f32x4 p = __builtin_nontemporal_load(&p4[i]);   // global_load_b128, TH=NT
    f32x4 t = __builtin_nontemporal_load(&t4[i]);
    a0 += masked_sq(t.x, p.x);
    a1 += masked_sq(t.y, p.y);
    a2 += masked_sq(t.z, p.z);
    a3 += masked_sq(t.w, p.w);
  }

  // Scalar tail (n % 4) handled once; re-converges before the WMMA reduction.
  if (blockIdx.x == 0 && threadIdx.x == 0) {
    for (long long i = nvec << 2; i < n; ++i)
      a0 += masked_sq(tgt[i], pred[i]);
  }

  // Wave32 reduction via matrix unit, then 8 waves -> 1 via LDS.
  float w = wave_sum_wmma(a0 + a2, a1 + a3);

  __shared__ float wave_sums[8];
  const int wave = threadIdx.x >> 5;
  const int lane = threadIdx.x & 31;
  if (lane == 0) wave_sums[wave] = w;
  __syncthreads();
  if (threadIdx.x == 0) {
    float s = 0.f;
    const int nwaves = blockDim.x >> 5;
    for (int i = 0; i < nwaves; ++i) s += wave_sums[i];   // fixed order: deterministic
    partials[blockIdx.x] = s;
  }
}

// Kernel 2: single block folds the per-block partials into d_out[0].
__global__ void __launch_bounds__(256)
reduce_partials(const float* __restrict__ partials,
                float* __restrict__ out,
                int nblocks) {
  float acc = 0.f;
  for (int i = threadIdx.x; i < nblocks; i += blockDim.x)
    acc += partials[i];

  float w = wave_sum_wmma(acc, 0.0f);

  __shared__ float wave_sums[8];
  const int wave = threadIdx.x >> 5;
  const int lane = threadIdx.x & 31;
  if (lane == 0) wave_sums[wave] = w;
  __syncthreads();
  if (threadIdx.x == 0) {
    float s = 0.f;
    const int nwaves = blockDim.x >> 5;
    for (int i = 0; i < nwaves; ++i) s += wave_sums[i];
    out[0] = s;   // direct store: no pre-zero / atomics needed, deterministic
  }
}

extern "C" void kernel_launch(void* const* d_in, const int* in_sizes, int n_in,
                              void* d_out, int out_size, void* d_ws, size_t ws_size,
                              hipStream_t stream) {
  const float* pred = (const float*)d_in[0];
  const float* tgt  = (const float*)d_in[1];
  const long long n = (long long)in_sizes[0];   // 4096*8192 = 33,554,432

  const int threads = 256;                       // 8 wave32s per block
  int blocks = 2048;                             // 16K waves; ~16 vec4-pairs/thread
  if ((size_t)blocks * sizeof(float) > ws_size && ws_size >= sizeof(float))
    blocks = (int)(ws_size / sizeof(float));     // safety clamp on scratch
  float* partials = (float*)d_ws;

  masked_sse_partial<<<blocks, threads, 0, stream>>>(pred, tgt, partials, n);
  reduce_partials<<<1, threads, 0, stream>>>(partials, (float*)d_out, blocks);
}